// SpikingT_89678917141316
// MI455X (gfx1250) — compile-verified
//
#include <hip/hip_runtime.h>

// CDNA5 / gfx1250: wave32, f32 WMMA 16x16x4 implicit-GEMM 3x3 conv fused with
// LIF integrate-and-fire scan + spike enhancement (all T register-resident).
//
// - GEMM-K is tap-major (k = tap*256 + c_in): 3x3 tap loop fully unrolled,
//   padding mask hoisted per tap, channel loop uses constant-stride addressing
//   (folds into VMEM immediate offsets; loads clamped + cndmask'd, no branches).
// - Each wave computes M=32 (two 16-channel tiles) against shared B fragments:
//   16 loads per K-step feed 12 WMMAs (vs 14:6 before) -> less L2 traffic,
//   higher matrix-pipe utilization.

typedef __attribute__((ext_vector_type(2))) float v2f;
typedef __attribute__((ext_vector_type(8))) float v8f;

namespace {
constexpr int Bn  = 16;         // batch
constexpr int Tn  = 6;          // timesteps
constexpr int Cn  = 256;        // channels
constexpr int HWn = 28;         // spatial dim
constexpr int Pn  = HWn * HWn;  // 784 pixels
constexpr int Kn  = Cn * 9;     // 2304 GEMM-K
constexpr int MT  = Cn / 32;    // 8 channel-pair tiles (M=32 per wave)
constexpr int PT  = Pn / 16;    // 49 pixel tiles (784 = 49*16 exact)
constexpr float NEG_EPS = -0.001f;
}

__global__ __launch_bounds__(256)
void spiking_conv_lif_wmma(const float* __restrict__ x,     // [B,T,C,H,W]
                           const float* __restrict__ Wt,    // [C,C,3,3]
                           const float* __restrict__ bias,  // [C]
                           const float* __restrict__ act_alpha,
                           float* __restrict__ out)         // [B,T,C,H,W]
{
    const int lane = threadIdx.x & 31;
    const int wave = blockIdx.x * (blockDim.x >> 5) + (threadIdx.x >> 5);

    // wave task: (b, 32-channel tile, pixel tile); this wave owns all T
    int tmp = wave;
    const int b  = tmp / (MT * PT); tmp -= b * (MT * PT);
    const int mt = tmp / PT;
    const int pt = tmp - mt * PT;
    if (b >= Bn) return;                    // uniform; exact grid

    const int m0 = mt * 32;
    const int p  = pt * 16 + (lane & 15);   // this lane's pixel (GEMM-N)
    const int h  = p / HWn;
    const int w  = p - h * HWn;
    const int kh = (lane >> 4) << 1;        // K sub-offset: 0 (lanes 0-15) / 2 (16-31)

    const float thr  = act_alpha[0];
    const int   mrow = m0 + (lane & 15);    // A rows: mrow (tile 0), mrow+16 (tile 1)

    const float* xb = x + (size_t)b * Tn * Cn * Pn;

    v8f accA[Tn], accB[Tn];
#pragma unroll
    for (int t = 0; t < Tn; ++t) {
        accA[t] = (v8f){0.f,0.f,0.f,0.f,0.f,0.f,0.f,0.f};
        accB[t] = (v8f){0.f,0.f,0.f,0.f,0.f,0.f,0.f,0.f};
    }

#pragma unroll
    for (int tap = 0; tap < 9; ++tap) {
        const int ky = tap / 3;             // compile-time after unroll
        const int kx = tap - ky * 3;

        // per-lane SAME-padding mask & clamped (always in-bounds) offset,
        // both loop-invariant for this tap
        const int  hh = h + ky - 1;
        const int  ww = w + kx - 1;
        const bool valid = ((unsigned)hh < (unsigned)HWn) &
                           ((unsigned)ww < (unsigned)HWn);
        const int  hs = hh < 0 ? 0 : (hh > HWn - 1 ? HWn - 1 : hh);
        const int  ws = ww < 0 ? 0 : (ww > HWn - 1 ? HWn - 1 : ww);
        const int  off = hs * HWn + ws + kh * Pn;   // includes lane-half K offset

        // A element for k = tap*256 + c + kh + j  ->  Wt[row*2304 + (c+kh+j)*9 + tap]
        const float* aptr0 = Wt + (size_t)mrow * Kn        + (size_t)kh * 9 + tap;
        const float* aptr1 = Wt + (size_t)(mrow + 16) * Kn + (size_t)kh * 9 + tap;

#pragma unroll 2
        for (int c0 = 0; c0 < Cn; c0 += 4) {
            v2f a0, a1;
            a0.x = aptr0[c0 * 9];           // K = tap*256 + c0 + kh
            a0.y = aptr0[c0 * 9 + 9];       // K = tap*256 + c0 + kh + 1
            a1.x = aptr1[c0 * 9];
            a1.y = aptr1[c0 * 9 + 9];

#pragma unroll
            for (int t = 0; t < Tn; ++t) {
                const float* xt = xb + (size_t)t * Cn * Pn + off;
                const float r0 = xt[c0 * Pn];        // clamped addr, always legal
                const float r1 = xt[c0 * Pn + Pn];
                v2f bf;
                bf.x = valid ? r0 : 0.0f;            // cndmask, no branch
                bf.y = valid ? r1 : 0.0f;
                accA[t] = __builtin_amdgcn_wmma_f32_16x16x4_f32(
                    false, a0, false, bf, (short)0, accA[t], false, false);
                accB[t] = __builtin_amdgcn_wmma_f32_16x16x4_f32(
                    false, a1, false, bf, (short)0, accB[t], false, false);
            }
        }
    }

    // ---- fused LIF scan + enhancement, register-resident ----
    // C/D layout: VGPR r -> channel row r (lanes 0-15) or r+8 (lanes 16-31)
    const int cbase = m0 + ((lane >> 4) << 3);

#pragma unroll
    for (int half = 0; half < 2; ++half) {
        v8f* acc = half ? accB : accA;

#pragma unroll
        for (int r = 0; r < 8; ++r) {
            const int   c  = cbase + half * 16 + r;
            const float bv = bias[c];

            float mem = 0.5f * thr;
            float ss  = 0.0f;
            float relusum = 0.0f;
#pragma unroll
            for (int t = 0; t < Tn; ++t) {
                float v = acc[t][r] + bv;                 // y = conv + bias
                relusum += fmaxf(v, 0.0f);
                mem += v;
                float sf = (mem >= thr) ? 1.0f : 0.0f;    // spike
                mem -= sf * thr;                           // reset
                ss  += sf;
                float inf = ((mem <= NEG_EPS) && (ss > 0.0f)) ? 1.0f : 0.0f;
                mem += inf * thr;                          // inhibit
                ss  -= inf;
                acc[t][r] = sf - inf;                      // emitted spike value
            }

            // enhancement: analog rate vs emitted spike count
            const int zfi = (int)(relusum / (float)Tn / thr * (float)Tn); // trunc
            int zi = (int)fmaxf((float)zfi - ss, 0.0f);
            zi = zi > Tn ? Tn : zi;

#pragma unroll
            for (int t = 0; t < Tn; ++t) {
                const float enh = (zi > (t + 1)) ? 1.0f : 0.0f;
                const float val = (acc[t][r] + enh) * thr;
                out[(size_t)(((b * Tn + t) * Cn + c) * Pn) + p] = val;
            }
        }
    }
}

extern "C" void kernel_launch(void* const* d_in, const int* in_sizes, int n_in,
                              void* d_out, int out_size, void* d_ws, size_t ws_size,
                              hipStream_t stream) {
    (void)in_sizes; (void)n_in; (void)d_ws; (void)ws_size; (void)out_size;
    const float* x    = (const float*)d_in[0];
    const float* Wt   = (const float*)d_in[1];
    const float* bias = (const float*)d_in[2];
    const float* aa   = (const float*)d_in[3];
    float* out        = (float*)d_out;

    constexpr int waves  = Bn * MT * PT;    // 6272 wave-tasks (M=32 each)
    constexpr int blocks = waves / 8;       // 8 wave32 waves per 256-thread block
    spiking_conv_lif_wmma<<<blocks, 256, 0, stream>>>(x, Wt, bias, aa, out);
}